// _NonLocalNd_46299747451316
// MI455X (gfx1250) — compile-verified
//
#include <hip/hip_runtime.h>
#include <hip/hip_bf16.h>

// ---------------------------------------------------------------------------
// NonLocal block (embedded gaussian) for MI455X / gfx1250.
// N=8, C=512, CI=256, H=W=56 -> L=3136.  ~107 GFLOP fp32, compute bound.
// All GEMMs use V_WMMA_F32_16X16X4_F32 (exact fp32). Attention is done
// flash-style (online softmax) so the 315MB L x L matrix is never stored.
// g / x tiles are DMA'd into double-buffered LDS by the Tensor Data Mover
// (TENSOR_LOAD_TO_LDS, TENSORcnt) so the copy overlaps the WMMA stream.
// ---------------------------------------------------------------------------

#define NB   8
#define CH   512
#define CI   256
#define LL   3136     // 56*56
#define NKT  196      // LL / 16 key tiles

typedef float v2f __attribute__((ext_vector_type(2)));
typedef float v8f __attribute__((ext_vector_type(8)));
typedef unsigned int u32;
typedef unsigned int u32x4 __attribute__((ext_vector_type(4)));
typedef int i32x4 __attribute__((ext_vector_type(4)));
typedef int i32x8 __attribute__((ext_vector_type(8)));

static __device__ __forceinline__ v8f wmma4(v2f a, v2f b, v8f c) {
  // D(16x16,f32) = A(16x4,f32) x B(4x16,f32) + C
  return __builtin_amdgcn_wmma_f32_16x16x4_f32(
      /*neg_a=*/false, a, /*neg_b=*/false, b,
      /*c_mod=*/(short)0, c, /*reuse_a=*/false, /*reuse_b=*/false);
}

static __device__ __forceinline__ v8f zero8() {
  v8f z = {0.f, 0.f, 0.f, 0.f, 0.f, 0.f, 0.f, 0.f};
  return z;
}

// ---------------------------------------------------------------------------
// Tensor Data Mover: DMA a 2D tile (tile_d1 rows x tile_d0 elems of f32,
// row pitch stride_d0 elems) from global into LDS at lds_byte_off, inserting
// pad_amt_code+1 dwords of LDS padding every 2^(pad_int_code+1) dwords.
// D# layout per CDNA5 ISA ch.8 (group0: count/lds/global addr, group1: dims).
// ---------------------------------------------------------------------------
static __device__ __forceinline__ void tdm_load_2d_f32(
    u32 lds_byte_off, const void* gptr, u32 tensor_d0, u32 tensor_d1,
    u32 tile_d0, u32 tile_d1, u32 stride_d0, u32 pad_int_code,
    u32 pad_amt_code) {
  unsigned long long ga = (unsigned long long)gptr;
  u32x4 g0;
  g0.x = 1u;                                   // count=1, user mode
  g0.y = lds_byte_off;                         // lds_addr
  g0.z = (u32)ga;                              // global_addr[31:0]
  g0.w = (u32)((ga >> 32) & 0x1FFFFFFull) | (2u << 30);  // addr[56:32]|type=2
  i32x8 g1;
  g1[0] = (int)((2u << 16) |                   // data_size = 4B
                (1u << 20) |                   // pad_enable
                (pad_int_code << 22) | (pad_amt_code << 25));
  g1[1] = (int)((tensor_d0 & 0xFFFFu) << 16);                 // dim0[15:0]
  g1[2] = (int)(((tensor_d0 >> 16) & 0xFFFFu) |               // dim0[31:16]
                ((tensor_d1 & 0xFFFFu) << 16));               // dim1[15:0]
  g1[3] = (int)(((tensor_d1 >> 16) & 0xFFFFu) | (tile_d0 << 16));
  g1[4] = (int)(tile_d1 & 0xFFFFu);            // tile_dim1, tile_dim2=0
  g1[5] = (int)stride_d0;                      // dim0_stride[31:0]
  g1[6] = 0;                                   // dim0_stride[47:32], dim1_str
  g1[7] = 0;
  i32x4 z4 = {0, 0, 0, 0};
#if defined(__clang_major__) && __clang_major__ >= 23
  i32x8 z8 = {0, 0, 0, 0, 0, 0, 0, 0};
  __builtin_amdgcn_tensor_load_to_lds(g0, g1, z4, z4, z8, 0);
#else
  __builtin_amdgcn_tensor_load_to_lds(g0, g1, z4, z4, 0);
#endif
}

static __device__ __forceinline__ u32 lds_off_of(const void* p) {
  // generic pointer into LDS aperture: low 32 bits are the LDS byte offset
  return (u32)(unsigned long long)p;
}

// ---------------------------------------------------------------------------
// Kernel 1: projection GEMM.  out[o,l] = sum_c W[o,c] * x[n,c,l] + bias[o]
// O = 256, K = 512.  Block = 256 thr (8 waves), block tile 128(M) x 64(N),
// each wave computes a 16 x 64 tile (4 wmma accumulators).
// x tile (16k x 64l) is TDM-DMA'd into double-buffered LDS [k][l] (stride 72,
// conflict-free) so the copy of tile i+1 overlaps the 128 WMMAs of tile i.
// If trans!=0, output is stored (n, L, CI) (used for theta so the attention
// kernel gets contiguous A-fragments).
// ---------------------------------------------------------------------------
__global__ __launch_bounds__(256) void nl_proj_kernel(
    const float* __restrict__ x, const float* __restrict__ W,
    const float* __restrict__ bias, float* __restrict__ out, int trans) {
  __shared__ float Xl[2 * 16 * 72];  // double-buffered [k][l], stride 72

  const int n  = blockIdx.z;
  const int l0 = blockIdx.y * 64;
  const int t  = threadIdx.x;
  const int wave = t >> 5, lane = t & 31, lo = lane & 15, hi = lane >> 4;
  const int m0 = blockIdx.x * 128 + wave * 16;

  const float* xb = x + (size_t)n * CH * LL;

  v8f acc[4];
#pragma unroll
  for (int nt = 0; nt < 4; ++nt) acc[nt] = zero8();

  // prologue: DMA tile 0 into buffer 0
  if (wave == 0)
    tdm_load_2d_f32(lds_off_of(&Xl[0]), xb + l0, LL, CH, 64, 16, LL,
                    /*pad: 8 dw every 64 dw ->*/ 5, 7);

#pragma unroll 1
  for (int it = 0; it < CH / 16; ++it) {
    if (wave == 0) __builtin_amdgcn_s_wait_tensorcnt(0);
    __syncthreads();  // publish buf[it&1]; buf[(it+1)&1] free for DMA
    if (wave == 0 && it + 1 < CH / 16)
      tdm_load_2d_f32(lds_off_of(&Xl[((it + 1) & 1) * (16 * 72)]),
                      xb + (size_t)(it + 1) * 16 * LL + l0, LL, CH, 64, 16,
                      LL, 5, 7);
    const float* Xb = &Xl[(it & 1) * (16 * 72)];
    const int k0 = it * 16;
#pragma unroll
    for (int kk = 0; kk < 4; ++kk) {
      v2f a = *(const v2f*)(W + (size_t)(m0 + lo) * CH + k0 + kk * 4 + 2 * hi);
      const int kb = kk * 4 + 2 * hi;
#pragma unroll
      for (int nt = 0; nt < 4; ++nt) {
        v2f b;
        b.x = Xb[kb * 72 + nt * 16 + lo];
        b.y = Xb[(kb + 1) * 72 + nt * 16 + lo];
        acc[nt] = wmma4(a, b, acc[nt]);
      }
    }
  }

#pragma unroll
  for (int nt = 0; nt < 4; ++nt) {
#pragma unroll
    for (int r = 0; r < 8; ++r) {
      int o = m0 + r + 8 * hi;
      int l = l0 + nt * 16 + lo;
      float v = acc[nt][r] + bias[o];
      if (trans)
        out[((size_t)n * LL + l) * CI + o] = v;     // (n, L, CI)
      else
        out[((size_t)n * CI + o) * LL + l] = v;     // (n, CI, L)
    }
  }
}

// ---------------------------------------------------------------------------
// Kernel 2: flash attention.  Per wave: 16 queries, loop over 196 key tiles.
//   S(16x16)  = theta(16x256) x phi(256x16)   (64 wmma, acc in v8f)
//   online softmax per row (rows live in fixed VGPR, 16 keys across lanes)
//   Y(16x256) += P(16x16) x g(16x256)         (64 wmma, acc in 16 v8f)
// Block = 128 thr (4 waves, 4 query tiles) sharing the phi/g key tiles in LDS.
// The g tile (256x16, LDS stride 20 = conflict-free) is TDM-DMA'd into a
// double buffer so tile kt+1 streams in during the 128 WMMAs of tile kt; the
// phi tile needs a transpose so it stays on the manual (VGPR) path.
// Output y stored transposed (n, L, CI) for the final projection.
// ---------------------------------------------------------------------------
__global__ __launch_bounds__(128) void nl_attn_kernel(
    const float* __restrict__ thetaT,  // (n, L, CI)
    const float* __restrict__ phi,     // (n, CI, L)
    const float* __restrict__ g,       // (n, CI, L)
    float* __restrict__ yT) {          // (n, L, CI)
  __shared__ float Pl[16 * 258];       // phi tile transposed: [key][c]
  __shared__ float Gl[2 * 256 * 20];   // g tile x2: [c][key], stride 20
  __shared__ float Pst[4 * 16 * 18];   // per-wave P staging: [q][key]

  const int n  = blockIdx.x / 49;
  const int qb = blockIdx.x % 49;
  const int t  = threadIdx.x;
  const int wave = t >> 5, lane = t & 31, lo = lane & 15, hi = lane >> 4;
  const int q0 = (qb * 4 + wave) * 16;

  const float* thb = thetaT + (size_t)n * LL * CI;
  const float* phb = phi + (size_t)n * CI * LL;
  const float* gb  = g + (size_t)n * CI * LL;
  float* Pw = Pst + wave * 16 * 18;

  v8f   yacc[16];
  float m[8], lsum[8];
#pragma unroll
  for (int nt = 0; nt < 16; ++nt) yacc[nt] = zero8();
#pragma unroll
  for (int r = 0; r < 8; ++r) { m[r] = -3.0e38f; lsum[r] = 0.f; }

  // prologue: DMA g tile 0 into buffer 0
  if (wave == 0)
    tdm_load_2d_f32(lds_off_of(&Gl[0]), gb, LL, CI, 16, 256, LL,
                    /*pad: 4 dw every 16 dw ->*/ 3, 3);

#pragma unroll 1
  for (int kt = 0; kt < NKT; ++kt) {
    const int k0 = kt * 16;
    __syncthreads();  // previous compute done; Pl and g buffers reusable
    // ---- manual transposed fill of the phi tile ----
#pragma unroll 4
    for (int j = 0; j < 32; ++j) {        // 4096 dwords, phi -> Pl[key][c]
      int idx = j * 128 + t;
      int c = idx >> 4, kk = idx & 15;
      Pl[kk * 258 + c] = phb[(size_t)c * LL + k0 + kk];
    }
    // prefetch next phi tile (global_prefetch_b8); g is handled by the TDM
    if (kt + 1 < NKT) {
      __builtin_prefetch(phb + (size_t)(t * 2 + 0) * LL + k0 + 16, 0, 0);
      __builtin_prefetch(phb + (size_t)(t * 2 + 1) * LL + k0 + 16, 0, 0);
    }
    if (wave == 0) __builtin_amdgcn_s_wait_tensorcnt(0);
    __syncthreads();  // publish Pl + Gl[kt&1]
    if (wave == 0 && kt + 1 < NKT)
      tdm_load_2d_f32(lds_off_of(&Gl[((kt + 1) & 1) * (256 * 20)]),
                      gb + k0 + 16, LL, CI, 16, 256, LL, 3, 3);
    const float* Gb = &Gl[(kt & 1) * (256 * 20)];

    // ---- S = theta x phi_tile (16x16 f32 accumulator) ----
    v8f s = zero8();
#pragma unroll
    for (int kk = 0; kk < 64; ++kk) {
      v2f a = *(const v2f*)(thb + (size_t)(q0 + lo) * CI + kk * 4 + 2 * hi);
      v2f b = *(const v2f*)(Pl + lo * 258 + kk * 4 + 2 * hi);
      s = wmma4(a, b, s);
    }

    // ---- online softmax update (rows: q = r + 8*hi, keys across 16 lanes) --
    float p[8], alpha[8];
#pragma unroll
    for (int r = 0; r < 8; ++r) {
      float sv = s[r] * 0.0625f;  // 1/sqrt(CI)
      float rm = sv;
      rm = fmaxf(rm, __shfl_xor(rm, 1));
      rm = fmaxf(rm, __shfl_xor(rm, 2));
      rm = fmaxf(rm, __shfl_xor(rm, 4));
      rm = fmaxf(rm, __shfl_xor(rm, 8));
      float mn = fmaxf(m[r], rm);
      alpha[r] = __expf(m[r] - mn);
      m[r] = mn;
      float pv = __expf(sv - mn);
      p[r] = pv;
      float rs = pv;
      rs += __shfl_xor(rs, 1);
      rs += __shfl_xor(rs, 2);
      rs += __shfl_xor(rs, 4);
      rs += __shfl_xor(rs, 8);
      lsum[r] = lsum[r] * alpha[r] + rs;
    }
#pragma unroll
    for (int nt = 0; nt < 16; ++nt)
#pragma unroll
      for (int r = 0; r < 8; ++r) yacc[nt][r] *= alpha[r];

    // ---- stage P through LDS to get A-fragment layout ----
#pragma unroll
    for (int r = 0; r < 8; ++r) Pw[(r + 8 * hi) * 18 + lo] = p[r];

    // ---- Y += P x g_tile ----
#pragma unroll
    for (int kk = 0; kk < 4; ++kk) {
      v2f a = *(const v2f*)(Pw + lo * 18 + kk * 4 + 2 * hi);
#pragma unroll
      for (int nt = 0; nt < 16; ++nt) {
        v2f b = *(const v2f*)(Gb + (nt * 16 + lo) * 20 + kk * 4 + 2 * hi);
        yacc[nt] = wmma4(a, b, yacc[nt]);
      }
    }
  }

  // ---- normalize and store y transposed (n, L, CI) ----
  float inv[8];
#pragma unroll
  for (int r = 0; r < 8; ++r) inv[r] = 1.0f / lsum[r];
#pragma unroll
  for (int nt = 0; nt < 16; ++nt)
#pragma unroll
    for (int r = 0; r < 8; ++r)
      yT[((size_t)n * LL + q0 + r + 8 * hi) * CI + nt * 16 + lo] =
          yacc[nt][r] * inv[r];
}

// ---------------------------------------------------------------------------
// Kernel 3: output projection + residual.
// out[n,o,l] = x[n,o,l] + b_out[o] + sum_c W_out[o,c] * y[n,c,l]
// y is stored (n, L, CI) so B-fragments are contiguous float2 global loads
// (L2-hot).  Block = 256 thr (8 waves), tile 128(M) x 64(N), K = 256.
// ---------------------------------------------------------------------------
__global__ __launch_bounds__(256) void nl_outproj_kernel(
    const float* __restrict__ yT, const float* __restrict__ Wo,
    const float* __restrict__ bo, const float* __restrict__ x,
    float* __restrict__ out) {
  const int n  = blockIdx.z;
  const int l0 = blockIdx.y * 64;
  const int t  = threadIdx.x;
  const int wave = t >> 5, lane = t & 31, lo = lane & 15, hi = lane >> 4;
  const int o0 = blockIdx.x * 128 + wave * 16;

  v8f acc[4];
#pragma unroll
  for (int nt = 0; nt < 4; ++nt) acc[nt] = zero8();

#pragma unroll 4
  for (int kk = 0; kk < 64; ++kk) {
    v2f a = *(const v2f*)(Wo + (size_t)(o0 + lo) * CI + kk * 4 + 2 * hi);
#pragma unroll
    for (int nt = 0; nt < 4; ++nt) {
      v2f b = *(const v2f*)(yT + ((size_t)n * LL + l0 + nt * 16 + lo) * CI +
                            kk * 4 + 2 * hi);
      acc[nt] = wmma4(a, b, acc[nt]);
    }
  }

#pragma unroll
  for (int nt = 0; nt < 4; ++nt) {
#pragma unroll
    for (int r = 0; r < 8; ++r) {
      int o = o0 + r + 8 * hi;
      int l = l0 + nt * 16 + lo;
      size_t off = ((size_t)n * CH + o) * LL + l;
      out[off] = acc[nt][r] + bo[o] + x[off];
    }
  }
}

// ---------------------------------------------------------------------------
extern "C" void kernel_launch(void* const* d_in, const int* in_sizes, int n_in,
                              void* d_out, int out_size, void* d_ws,
                              size_t ws_size, hipStream_t stream) {
  (void)in_sizes; (void)n_in; (void)out_size; (void)ws_size;

  const float* x       = (const float*)d_in[0];
  const float* W_g     = (const float*)d_in[1];
  const float* b_g     = (const float*)d_in[2];
  const float* W_theta = (const float*)d_in[3];
  const float* b_theta = (const float*)d_in[4];
  const float* W_phi   = (const float*)d_in[5];
  const float* b_phi   = (const float*)d_in[6];
  const float* W_out   = (const float*)d_in[7];
  const float* b_out   = (const float*)d_in[8];
  float* out = (float*)d_out;

  // workspace: 4 fp32 buffers of N*L*CI = 6,422,528 floats (~25.7 MB each)
  float* ws = (float*)d_ws;
  const size_t per = (size_t)NB * LL * CI;
  float* thetaT = ws;            // (n, L, CI)
  float* phiB   = ws + per;      // (n, CI, L)
  float* gB     = ws + 2 * per;  // (n, CI, L)
  float* yT     = ws + 3 * per;  // (n, L, CI)

  dim3 pgrid(CI / 128, LL / 64, NB);  // (2, 49, 8)
  nl_proj_kernel<<<pgrid, 256, 0, stream>>>(x, W_theta, b_theta, thetaT, 1);
  nl_proj_kernel<<<pgrid, 256, 0, stream>>>(x, W_phi, b_phi, phiB, 0);
  nl_proj_kernel<<<pgrid, 256, 0, stream>>>(x, W_g, b_g, gB, 0);

  nl_attn_kernel<<<dim3(NB * 49), 128, 0, stream>>>(thetaT, phiB, gB, yT);

  dim3 ogrid(CH / 128, LL / 64, NB);  // (4, 49, 8)
  nl_outproj_kernel<<<ogrid, 256, 0, stream>>>(yT, W_out, b_out, x, out);
}